// InterClassContrastiveLoss_19696720019941
// MI455X (gfx1250) — compile-verified
//
#include <hip/hip_runtime.h>
#include <hip/hip_bf16.h>
#include <math.h>
#include <stdint.h>

#define B_SZ 8192
#define D_SZ 512
#define P_SZ 2
#define N_SZ 2
#define INV_T 10.0f

#define CHUNK 64               // K-chunk in floats (256 B per row)
#define RS    68               // LDS row stride: CHUNK + 4 pad (8B aligned, bank-spread)
#define NROW  80               // 16 anchors + 64 partners
#define NCHUNK (D_SZ / CHUNK)  // 8

typedef float v2f __attribute__((ext_vector_type(2)));
typedef float v8f __attribute__((ext_vector_type(8)));

__device__ __forceinline__ float softplusf(float x) {
  // numerically stable softplus: max(x,0) + log1p(exp(-|x|))
  return fmaxf(x, 0.0f) + log1pf(expf(-fabsf(x)));
}

__global__ void init_ws_kernel(float* ws) {
  ws[0] = 0.0f;  // loss accumulator
  ws[1] = 0.0f;  // valid-anchor count
}

// One wave per row: rnorm[row] = rsqrt(max(sum(f^2), 1e-24))
__global__ __launch_bounds__(256) void rnorm_kernel(const float* __restrict__ feat,
                                                    float* __restrict__ rnorm) {
  const int wave = threadIdx.x >> 5;
  const int lane = threadIdx.x & 31;
  const int row  = blockIdx.x * 8 + wave;
  const float4* fr = (const float4*)(feat + (size_t)row * D_SZ);
  float s = 0.0f;
#pragma unroll
  for (int j = 0; j < 4; ++j) {
    float4 v = fr[lane + 32 * j];
    s += v.x * v.x + v.y * v.y + v.z * v.z + v.w * v.w;
  }
#pragma unroll
  for (int off = 16; off > 0; off >>= 1) s += __shfl_xor(s, off, 32);
  if (lane == 0) rnorm[row] = rsqrtf(fmaxf(s, 1e-24f));
}

// One wave (block of 32) handles 16 anchors.
// LDS rows: [0..15] = anchors, [16+16*s+n] = slot-s partner of anchor n.
// 4 WMMA accumulator tiles (one per slot); tile diagonal (m,m) is
// anchor m dot its slot-s partner. Staging uses async global->LDS DMA,
// double-buffered across K-chunks.
__global__ __launch_bounds__(32) void pair_loss_kernel(
    const float* __restrict__ feat,
    const int* __restrict__ pos_idx,
    const int* __restrict__ neg_idx,
    const unsigned char* __restrict__ valid,
    const float* __restrict__ rnorm,
    float* __restrict__ ws_acc) {
  __shared__ float rowbuf[2][NROW * RS];
  __shared__ int   rowidx[NROW];
  __shared__ float tilebuf[16 * 16];

  const int lane = threadIdx.x;       // 0..31
  const int base = blockIdx.x * 16;   // first anchor of this wave
  const int half = lane >> 4;         // K-pair / row-pair selector
  const int lm   = lane & 15;         // matrix row / column index

  // Build the 80 gather row indices (divergent, pre-WMMA so EXEC recovers).
  for (int r = lane; r < NROW; r += 32) {
    int idx;
    if (r < 16) {
      idx = base + r;
    } else {
      int s = (r - 16) >> 4;   // slot 0..3
      int n = (r - 16) & 15;   // anchor within group
      int a = base + n;
      idx = (s < P_SZ) ? pos_idx[a * P_SZ + s]
                       : neg_idx[a * N_SZ + (s - P_SZ)];
    }
    rowidx[r] = idx;
  }
  __syncthreads();

  // Issue one K-chunk of async global->LDS copies. One b128 per lane covers
  // 16 B; lanes 0-15 service row 2j, lanes 16-31 service row 2j+1, so each
  // instruction moves two full 256 B row-chunks. 40 async ops per chunk.
  auto issue_chunk = [&](int c) {
    const int koff = c * CHUNK;
    float* dstbase = &rowbuf[c & 1][0];
    // compute reads of this buffer (2 chunks ago) are fully drained
    asm volatile("s_wait_dscnt 0x0" ::: "memory");
    for (int jj = 0; jj < NROW / 2; jj += 8) {
      int ridx[8];
#pragma unroll
      for (int t = 0; t < 8; ++t)
        ridx[t] = rowidx[2 * (jj + t) + half];   // batched ds reads, one wait
#pragma unroll
      for (int t = 0; t < 8; ++t) {
        const int row = 2 * (jj + t) + half;
        const float* g = feat + (size_t)ridx[t] * D_SZ + koff + lm * 4;
        uint32_t laddr = (uint32_t)(uintptr_t)(dstbase + row * RS + lm * 4);
        asm volatile("global_load_async_to_lds_b128 %0, %1, off"
                     :: "v"(laddr), "v"(g)
                     : "memory");
      }
    }
  };

  v8f acc[4];
#pragma unroll
  for (int s = 0; s < 4; ++s)
    acc[s] = (v8f){0.f, 0.f, 0.f, 0.f, 0.f, 0.f, 0.f, 0.f};

  issue_chunk(0);
  for (int c = 0; c < NCHUNK; ++c) {
    if (c + 1 < NCHUNK) {
      issue_chunk(c + 1);
      // async loads complete in order: <=40 outstanding => chunk c landed
      asm volatile("s_wait_asynccnt %0" :: "i"(NROW / 2) : "memory");
    } else {
      asm volatile("s_wait_asynccnt 0x0" ::: "memory");
    }
    const float* buf = &rowbuf[c & 1][0];

    // WMMA K loop: V_WMMA_F32_16X16X4_F32, uniform EXEC throughout.
    // A frag (16x4 f32): lane<16 holds A[lm][k..k+1], lane>=16 A[lm][k+2..k+3]
    // B frag (4x16 f32): lane<16 holds B[k..k+1][lm],  lane>=16 B[k+2..k+3][lm]
    //   with B[k][n] = feat[partner_n][k]  -> contiguous b64 LDS reads.
#pragma unroll 4
    for (int k = 0; k < CHUNK; k += 4) {
      v2f a = *(const v2f*)&buf[lm * RS + k + 2 * half];
#pragma unroll
      for (int s = 0; s < 4; ++s) {
        v2f b = *(const v2f*)&buf[(16 + 16 * s + lm) * RS + k + 2 * half];
        acc[s] = __builtin_amdgcn_wmma_f32_16x16x4_f32(
            false, a, false, b, (short)0, acc[s], false, false);
      }
    }
  }

  // Extract tile diagonals via LDS: C/D layout is VGPR v: (M=v+8*half, N=lm).
  float dots[4];
#pragma unroll
  for (int s = 0; s < 4; ++s) {
    __syncthreads();
#pragma unroll
    for (int v = 0; v < 8; ++v)
      tilebuf[(v + 8 * half) * 16 + lm] = acc[s][v];
    __syncthreads();
    dots[s] = tilebuf[lm * 17];  // (m,m); meaningful for lanes 0..15
  }

  float lsum = 0.0f, vsum = 0.0f;
  if (lane < 16) {
    const int a = base + lane;
    const float rni = rnorm[a];
    float simv[4];
#pragma unroll
    for (int s = 0; s < 4; ++s) {
      int pidx = rowidx[16 + 16 * s + lane];
      simv[s] = dots[s] * rni * rnorm[pidx] * INV_T;
    }
    const float vm = valid[a] ? 1.0f : 0.0f;
    float l = 0.0f;
#pragma unroll
    for (int p = 0; p < 2; ++p)
#pragma unroll
      for (int n = 2; n < 4; ++n)
        l += softplusf(simv[n] - simv[p]);
    lsum = l * vm;
    vsum = vm;
  }
#pragma unroll
  for (int off = 16; off > 0; off >>= 1) {
    lsum += __shfl_xor(lsum, off, 32);
    vsum += __shfl_xor(vsum, off, 32);
  }
  if (lane == 0) {
    atomicAdd(&ws_acc[0], lsum);
    atomicAdd(&ws_acc[1], vsum);
  }
}

__global__ void finalize_kernel(const float* __restrict__ ws, float* __restrict__ out) {
  const float cnt = ws[1];
  const float np  = cnt * (float)(P_SZ * N_SZ);
  out[0] = (np > 0.0f) ? ws[0] / np : 0.0f;
}

extern "C" void kernel_launch(void* const* d_in, const int* in_sizes, int n_in,
                              void* d_out, int out_size, void* d_ws, size_t ws_size,
                              hipStream_t stream) {
  const float*         feat    = (const float*)d_in[0];
  const int*           pos_idx = (const int*)d_in[1];
  const int*           neg_idx = (const int*)d_in[2];
  const unsigned char* valid   = (const unsigned char*)d_in[3];

  float* ws    = (float*)d_ws;
  float* rnorm = ws + 8;  // 8192 floats of scratch for inverse norms

  init_ws_kernel<<<1, 1, 0, stream>>>(ws);
  rnorm_kernel<<<B_SZ / 8, 256, 0, stream>>>(feat, rnorm);
  pair_loss_kernel<<<B_SZ / 16, 32, 0, stream>>>(feat, pos_idx, neg_idx, valid,
                                                 rnorm, ws);
  finalize_kernel<<<1, 1, 0, stream>>>(ws, (float*)d_out);
}